// FMFMSynapticNeuron_42262478192734
// MI455X (gfx1250) — compile-verified
//
#include <hip/hip_runtime.h>
#include <cstdint>

// SNN Synaptic neuron scan: out[t,b] = spike step over T=50, B=262144.
// Bandwidth-bound (157 MB once-touched, ~6.7 us floor @ 23.3 TB/s).
// CDNA5 data path: deep async global->LDS prefetch pipeline (ASYNCcnt).

#define T_STEPS 50
#define DEPTH   8        // async prefetch depth (timesteps in flight per wave)
#define BLOCK   256

// global_load_async_to_lds_b64: VDST = per-lane LDS byte offset, VADDR = 64-bit
// per-lane global address, no SADDR ("off"). Tracked by ASYNCcnt (in-order).
__device__ __forceinline__ void async_copy_b64(unsigned lds_off,
                                               unsigned long long gaddr) {
  asm volatile("global_load_async_to_lds_b64 %0, %1, off"
               :
               : "v"(lds_off), "v"(gaddr)
               : "memory");
}

__device__ __forceinline__ void wait_async_le7() {
  asm volatile("s_wait_asynccnt 0x7" ::: "memory");
}
__device__ __forceinline__ void wait_async_le0() {
  asm volatile("s_wait_asynccnt 0x0" ::: "memory");
}

__global__ __launch_bounds__(BLOCK)
void snn_synaptic_scan(const float* __restrict__ x,   // [T, B, 2]
                       const float* __restrict__ W,   // [1, 2]
                       float* __restrict__ out,       // [T, B, 1]
                       int B) {
  __shared__ float2 buf[DEPTH][BLOCK];   // 16 KB circular buffer

  const int tid = threadIdx.x;
  const long long b = (long long)blockIdx.x * BLOCK + tid;
  if (b >= B) return;

  // Uniform weights -> scalar loads into SGPRs.
  const float w0 = W[0];
  const float w1 = W[1];

  // Per-lane source: element (t=0, b); stride between timesteps is B float2s.
  const float2* src = (const float2*)x + b;

  // LDS byte offset of this lane's slot in buffer 0 (generic ptr low 32 bits
  // == LDS offset on AMDGPU). Buffer stride = BLOCK*8 = 2048 bytes.
  const unsigned lds_base = (unsigned)(uintptr_t)&buf[0][tid];

  // Prologue: fill the pipeline with DEPTH in-flight timesteps.
  #pragma unroll
  for (int i = 0; i < DEPTH; ++i)
    async_copy_b64(lds_base + i * (BLOCK * 8),
                   (unsigned long long)(uintptr_t)(src + (long long)i * B));

  float syn = 0.0f, mem = 0.0f;

  // Steady state: DEPTH loads outstanding; asynccnt<=DEPTH-1 retires load t.
  for (int t = 0; t < T_STEPS - DEPTH; ++t) {
    wait_async_le7();
    float2 s = buf[t & (DEPTH - 1)][tid];          // ds_load_b64

    float cur   = s.x * w0 + s.y * w1;
    float reset = (mem > 1.0f) ? 1.0f : 0.0f;      // THRESHOLD = 1
    syn = 0.9f * syn + cur;                        // ALPHA
    mem = 0.9f * mem + syn - reset;                // BETA, reset*THRESHOLD
    float spk = (mem > 1.0f) ? 1.0f : 0.0f;
    __builtin_nontemporal_store(spk, out + (long long)t * B + b);

    // Recycle the slot we just consumed for timestep t+DEPTH.
    async_copy_b64(lds_base + (t & (DEPTH - 1)) * (BLOCK * 8),
                   (unsigned long long)(uintptr_t)(src + (long long)(t + DEPTH) * B));
  }

  // Tail: everything already issued; drain with a single full wait.
  wait_async_le0();
  #pragma unroll
  for (int t = T_STEPS - DEPTH; t < T_STEPS; ++t) {
    float2 s = buf[t & (DEPTH - 1)][tid];
    float cur   = s.x * w0 + s.y * w1;
    float reset = (mem > 1.0f) ? 1.0f : 0.0f;
    syn = 0.9f * syn + cur;
    mem = 0.9f * mem + syn - reset;
    float spk = (mem > 1.0f) ? 1.0f : 0.0f;
    __builtin_nontemporal_store(spk, out + (long long)t * B + b);
  }
}

extern "C" void kernel_launch(void* const* d_in, const int* in_sizes, int n_in,
                              void* d_out, int out_size, void* d_ws, size_t ws_size,
                              hipStream_t stream) {
  const float* x = (const float*)d_in[0];   // spike_seq [50, B, 2] f32
  const float* W = (const float*)d_in[1];   // [1, 2] f32
  float* out = (float*)d_out;               // [50, B, 1] f32

  const int B = out_size / T_STEPS;         // 262144
  dim3 grid((B + BLOCK - 1) / BLOCK), block(BLOCK);
  snn_synaptic_scan<<<grid, block, 0, stream>>>(x, W, out, B);
}